// SwitchGate_91096256348825
// MI455X (gfx1250) — compile-verified
//
#include <hip/hip_runtime.h>
#include <math.h>

// ---------------- problem constants ----------------
#define N_TOKENS   131072
#define D_MODEL    1024
#define NUM_EXPERT 64
#define CAPACITY   2458           // ceil(1.2 * 131072 / 64)
#define CHUNK      64             // K-chunk staged in LDS
#define ROWS_PER_BLOCK 128        // tokens per workgroup in gate kernel
#define NBLOCKS    (N_TOKENS / ROWS_PER_BLOCK)   // 1024

typedef __attribute__((ext_vector_type(2))) float v2f;
typedef __attribute__((ext_vector_type(8))) float v8f;

// ---------------- init: zero the global softmax column sums ----------------
__global__ void sg_init_kernel(float* __restrict__ scoreSum) {
    scoreSum[threadIdx.x] = 0.0f;
}

// ---------------- gate GEMM + softmax/argmax epilogue ----------------
// grid: NBLOCKS x 256 threads (8 waves); wave w handles tokens [base+16w, base+16w+16)
__global__ __launch_bounds__(256)
void sg_gate_kernel(const float* __restrict__ inp,    // [N, 1024]
                    const float* __restrict__ W,      // [1024, 64]
                    const float* __restrict__ bias,   // [64]
                    const float* __restrict__ noise,  // [N, 64]
                    float* __restrict__ out,          // [2N+1]; writes [N,2N)
                    int*   __restrict__ top1_idx,     // [N]
                    int*   __restrict__ blockHist,    // [NBLOCKS*64]
                    float* __restrict__ scoreSum)     // [64] (atomic)
{
    // A-tile: 128 rows x (64+4 pad) f32 = 34816 B, aliased afterwards with
    // per-wave score tiles: 8 waves x (16 x 65) f32 = 33280 B.
    __shared__ float lds[128 * 68];
    __shared__ float maxArr[128];   // per-wave 16 token maxes
    __shared__ float invArr[128];   // per-wave 16 token 1/denominators
    __shared__ float sumLds[64];    // block-level per-expert softmax sums
    __shared__ int   hist[64];      // block-level per-expert top1 counts

    const int tid   = threadIdx.x;
    const int lane  = tid & 31;
    const int w     = tid >> 5;                 // wave id 0..7
    const int blk   = blockIdx.x;
    const int rowBase = blk * ROWS_PER_BLOCK;   // first token of this block
    const int l15   = lane & 15;
    const int koff  = (lane >> 4) << 1;         // 0 (lanes 0-15) or 2 (lanes 16-31)

    v8f acc[4] = {};                            // 16x64 f32 accumulator (4 N-tiles)

    // A fragment source in LDS: row = wave-local token (l15), cols k4+koff, k4+koff+1
    const int aBase = (w * 16 + l15) * 68 + koff;
    const float* wcol = W + l15;                // per-lane column base into W

    for (int kc = 0; kc < D_MODEL; kc += CHUNK) {
        // ---- cooperative stage: inp[rowBase..+128][kc..kc+64) -> LDS (b128 loads)
        #pragma unroll
        for (int i = 0; i < 8; ++i) {
            int v   = tid + i * 256;            // float4 index in [0, 2048)
            int row = v >> 4;
            int c4  = (v & 15) << 2;
            float4 d = *(const float4*)(inp + (size_t)(rowBase + row) * D_MODEL + kc + c4);
            *(float4*)(&lds[row * 68 + c4]) = d;
        }
        __syncthreads();

        // ---- 16 K-steps of 4, 4 WMMAs each (N tiles of 16 experts)
        #pragma unroll 4
        for (int k4 = 0; k4 < CHUNK; k4 += 4) {
            v2f a = *(const v2f*)(&lds[aBase + k4]);      // A: VGPR0=K{0|2}, VGPR1=K{1|3}
            const int kg = kc + k4 + koff;
            const float* wr0 = wcol + (size_t)kg * NUM_EXPERT;
            const float* wr1 = wr0 + NUM_EXPERT;
            #pragma unroll
            for (int t = 0; t < 4; ++t) {
                v2f bf;
                bf.x = wr0[t * 16];                        // B: K{0|2}, N = t*16 + l15
                bf.y = wr1[t * 16];                        //    K{1|3}
                acc[t] = __builtin_amdgcn_wmma_f32_16x16x4_f32(
                    false, a, false, bf, (short)0, acc[t], false, false);
            }
        }
        __syncthreads();
    }

    // ================= epilogue =================
    if (tid < 64) { sumLds[tid] = 0.0f; hist[tid] = 0; }
    __syncthreads();

    // C/D layout: VGPR r, lanes 0-15 -> M=r, N=lane; lanes 16-31 -> M=r+8, N=lane-16
    const int scoreBase = w * 1040;             // 16 * 65 per wave
    #pragma unroll
    for (int t = 0; t < 4; ++t) {
        const int e  = t * 16 + l15;
        const float be = bias[e];
        #pragma unroll
        for (int r = 0; r < 8; ++r) {
            const int tr  = (lane < 16) ? r : (r + 8);    // wave-local token
            const int tok = rowBase + w * 16 + tr;
            float v = acc[t][r] + be
                    + noise[(size_t)tok * NUM_EXPERT + e] * 0.2f + 0.9f;
            lds[scoreBase + tr * 65 + e] = v;
        }
    }
    __syncthreads();

    // per-token max / argmax / softmax denominator (lanes 0-15: one token each)
    if (lane < 16) {
        const int tt = lane;
        const float* s = &lds[scoreBase + tt * 65];
        float mx = s[0]; int am = 0;
        for (int e = 1; e < 64; ++e) { float v = s[e]; if (v > mx) { mx = v; am = e; } }
        float den = 0.0f;
        for (int e = 0; e < 64; ++e) den += __expf(s[e] - mx);
        float inv = 1.0f / den;                 // softmax value at argmax
        maxArr[w * 16 + tt] = mx;
        invArr[w * 16 + tt] = inv;
        const int tok = rowBase + w * 16 + tt;
        out[N_TOKENS + tok] = inv;              // top1_score
        top1_idx[tok] = am;
        atomicAdd(&hist[am], 1);
    }
    __syncthreads();

    // per-expert softmax column partial sums (lane handles experts lane, lane+32)
    {
        float s0 = 0.0f, s1 = 0.0f;
        for (int tt = 0; tt < 16; ++tt) {
            float mx  = maxArr[w * 16 + tt];
            float inv = invArr[w * 16 + tt];
            s0 += __expf(lds[scoreBase + tt * 65 + lane]      - mx) * inv;
            s1 += __expf(lds[scoreBase + tt * 65 + lane + 32] - mx) * inv;
        }
        atomicAdd(&sumLds[lane],      s0);
        atomicAdd(&sumLds[lane + 32], s1);
    }
    __syncthreads();

    if (tid < 64) {
        atomicAdd(&scoreSum[tid], sumLds[tid]);
        blockHist[blk * 64 + tid] = hist[tid];
    }
}

// ---------------- per-expert exclusive prefix over block histograms ----------------
__global__ void sg_scan_kernel(const int* __restrict__ blockHist,
                               int* __restrict__ blockPrefix,
                               int* __restrict__ totalCount)
{
    const int e = threadIdx.x;                  // 64 threads
    int run = 0;
    for (int bb = 0; bb < NBLOCKS; ++bb) {
        int c = blockHist[bb * 64 + e];
        blockPrefix[bb * 64 + e] = run;
        run += c;
    }
    totalCount[e] = run;
}

// ---------------- arrival-order rank + capacity prune ----------------
__global__ __launch_bounds__(128)
void sg_rank_kernel(const int* __restrict__ top1_idx,
                    const int* __restrict__ blockPrefix,
                    float* __restrict__ out)     // writes [0, N)
{
    __shared__ int waveHist[4][64];
    const int tid  = threadIdx.x;
    const int lane = tid & 31;
    const int w    = tid >> 5;
    const int blk  = blockIdx.x;
    const int tok  = blk * 128 + tid;
    const int myE  = top1_idx[tok];

    unsigned myMask = 0u;
    for (int e = 0; e < 64; ++e) {
        unsigned bal = (unsigned)__ballot(myE == e);   // wave32: low 32 bits
        if (lane == 0) waveHist[w][e] = __popc(bal);
        if (myE == e)  myMask = bal;
    }
    __syncthreads();

    int rank = __popc(myMask & ((1u << lane) - 1u));   // lane order == token order
    #pragma unroll
    for (int ww = 0; ww < 4; ++ww)
        if (ww < w) rank += waveHist[ww][myE];
    rank += blockPrefix[blk * 64 + myE];

    out[tok] = (rank < CAPACITY) ? (float)myE : -1.0f;
}

// ---------------- load-balancing loss ----------------
__global__ void sg_loss_kernel(const int* __restrict__ totalCount,
                               const float* __restrict__ scoreSum,
                               float* __restrict__ out)  // writes [2N]
{
    __shared__ float cbuf[64];
    __shared__ float tbuf[64];
    const int e = threadIdx.x;
    float c = fminf((float)totalCount[e], (float)CAPACITY);
    cbuf[e] = c;
    tbuf[e] = c * scoreSum[e];
    __syncthreads();
    if (e == 0) {
        float nv = 0.0f, num = 0.0f;
        for (int i = 0; i < 64; ++i) { nv += cbuf[i]; num += tbuf[i]; }
        out[2 * N_TOKENS] = num / (nv * nv) * (float)NUM_EXPERT;
    }
}

// ---------------- launch ----------------
extern "C" void kernel_launch(void* const* d_in, const int* in_sizes, int n_in,
                              void* d_out, int out_size, void* d_ws, size_t ws_size,
                              hipStream_t stream)
{
    const float* inp   = (const float*)d_in[0];   // [N, 1024]
    const float* W     = (const float*)d_in[1];   // [1024, 64]
    const float* bias  = (const float*)d_in[2];   // [64]
    const float* noise = (const float*)d_in[3];   // [N, 64]
    float* out = (float*)d_out;                   // [N pruned | N score | 1 loss]

    char* ws = (char*)d_ws;
    size_t off = 0;
    int*   top1_idx    = (int*)(ws + off); off += (size_t)N_TOKENS * 4;
    int*   blockHist   = (int*)(ws + off); off += (size_t)NBLOCKS * 64 * 4;
    int*   blockPrefix = (int*)(ws + off); off += (size_t)NBLOCKS * 64 * 4;
    int*   totalCount  = (int*)(ws + off); off += 256;
    float* scoreSum    = (float*)(ws + off); off += 256;
    (void)ws_size; (void)in_sizes; (void)n_in; (void)out_size;

    hipLaunchKernelGGL(sg_init_kernel, dim3(1), dim3(64), 0, stream, scoreSum);
    hipLaunchKernelGGL(sg_gate_kernel, dim3(NBLOCKS), dim3(256), 0, stream,
                       inp, W, bias, noise, out, top1_idx, blockHist, scoreSum);
    hipLaunchKernelGGL(sg_scan_kernel, dim3(1), dim3(64), 0, stream,
                       blockHist, blockPrefix, totalCount);
    hipLaunchKernelGGL(sg_rank_kernel, dim3(NBLOCKS), dim3(128), 0, stream,
                       top1_idx, blockPrefix, out);
    hipLaunchKernelGGL(sg_loss_kernel, dim3(1), dim3(64), 0, stream,
                       totalCount, scoreSum, out);
}